// MultiomicTransformer_4801773437357
// MI455X (gfx1250) — compile-verified
//
#include <hip/hip_runtime.h>
#include <math.h>

// ---------------------------------------------------------------------------
// MultiomicTransformer forward for MI455X (gfx1250), fp32 WMMA everywhere,
// async global->LDS (GLOBAL_LOAD_ASYNC_TO_LDS_B128) double-buffered staging.
//
// Input pointer order assumption: setup_inputs() dict flattened in insertion
// order, recursively (params dict in creation order):
//   0 atac_windows, 1 tf_expr, 2 bias,
//   3 tf_id_emb, 4 tg_q_emb, 5 tg_id_emb,
//   6 tfin_w1, 7 tfin_w2, 8 tfin_g, 9 tfin_b,
//   10 atacin_w1, 11 atacin_w2, 12 atacin_g, 13 atacin_b,
//   14 pool_q1, 15 pool_q2, 16 pooled_w1, 17 pooled_w2, 18 pooled_g,
//   19 pooled_b, 20 pred_w1, 21 pred_w2, 22 shortcut_scale,
//   23+12l enc[l]{wqkv,bqkv,wo,bo,w1,b1,w2,b2,g1,b1n,g2,b2n} (l=0..3),
//   71 c_tf2atac{wq,bq,wk,bk,wv,bv,wo,bo,g,b}, 81 c_atac2tf, 91 c_tg2atac,
//   101 tf_ids (int), 102 tg_ids (int)
// Workspace requirement: ~196 MB of float scratch (bump allocated below).
// ---------------------------------------------------------------------------

#define DD 512
#define HH 8
#define DKK 64
#define FFD 2048
#define BB 8
#define WW 1024
#define TT 512
#define GG 1024

typedef __attribute__((ext_vector_type(2))) float v2f;
typedef __attribute__((ext_vector_type(8))) float v8f;

__device__ inline v8f wmma4(v2f a, v2f b, v8f c) {
  // D = A(16x4 f32) x B(4x16 f32) + C(16x16 f32)
  return __builtin_amdgcn_wmma_f32_16x16x4_f32(false, a, false, b, (short)0, c,
                                               false, false);
}

// Async 16B copy global -> LDS (per lane). ASYNCcnt-tracked; no VGPR data.
// Low 32 bits of a flat LDS pointer == LDS byte offset (ISA 10.2).
__device__ inline void async_cp16(void* lds, const void* g) {
  unsigned la = (unsigned)(size_t)lds;
  unsigned long long ga = (unsigned long long)(size_t)g;
  asm volatile("global_load_async_to_lds_b128 %0, %1, off" ::"v"(la), "v"(ga)
               : "memory");
}
__device__ inline void async_wait0() {
  asm volatile("s_wait_asynccnt 0x0" ::: "memory");
}

__device__ inline float wsum32(float v) {
#pragma unroll
  for (int o = 16; o > 0; o >>= 1) v += __shfl_xor(v, o, 32);
  return v;
}
__device__ inline float wmax32(float v) {
#pragma unroll
  for (int o = 16; o > 0; o >>= 1) v = fmaxf(v, __shfl_xor(v, o, 32));
  return v;
}
__device__ inline float hmax16(float v) {
#pragma unroll
  for (int o = 8; o > 0; o >>= 1) v = fmaxf(v, __shfl_xor(v, o, 16));
  return v;
}
__device__ inline float hsum16(float v) {
#pragma unroll
  for (int o = 8; o > 0; o >>= 1) v += __shfl_xor(v, o, 16);
  return v;
}

// ---------------------------------------------------------------------------
// Generic fp32 WMMA GEMM: C = act(alpha*(A@B) + bias + resid)
// A: MxK row-major (lda=K). B: KxN row-major, or NxK if TRANSB (ldb).
// Block = 256 threads = 8 waves; block tile 64x128; wave tile 32x32.
// Double-buffered async LDS staging; one barrier per 16-deep K-step.
// All M%64==0, N%128==0, K%16==0 in this model.
// act: 0=none, 1=relu, 2=clamp +-50
// ---------------------------------------------------------------------------
template <bool TRANSB>
__global__ __launch_bounds__(256) void k_gemm(
    const float* __restrict__ A, const float* __restrict__ B,
    const float* __restrict__ bias, const float* __restrict__ resid,
    float* __restrict__ C, int M, int N, int K, int lda, int ldb, int ldc,
    float alpha, int act) {
  __shared__ alignas(16) float As[2][64][20];
  constexpr int BR = TRANSB ? 128 : 16;
  constexpr int BC = TRANSB ? 20 : 128;
  __shared__ alignas(16) float Bs[2][BR][BC];

  const int tid = threadIdx.x;
  const int lane = tid & 31, wv = tid >> 5;
  const int wm = wv >> 2, wn = wv & 3;  // 2 x 4 wave grid
  const int hi = (lane >> 4) & 1;
  const int ln16 = lane & 15;
  const long tm = (long)blockIdx.y * 64;
  const long tn = (long)blockIdx.x * 128;

  v8f vz = {0.f, 0.f, 0.f, 0.f, 0.f, 0.f, 0.f, 0.f};
  v8f acc[2][2];
#pragma unroll
  for (int i = 0; i < 2; i++)
#pragma unroll
    for (int j = 0; j < 2; j++) acc[i][j] = vz;

  auto stage = [&](int buf, int k0) {
    {  // A tile 64x16: one 16B async per thread
      int r = tid >> 2, c4 = (tid & 3) << 2;
      async_cp16(&As[buf][r][c4], A + (tm + r) * (long)lda + k0 + c4);
    }
    if constexpr (TRANSB) {  // B stored NxK: tile 128 rows x 16 cols
#pragma unroll
      for (int rr = 0; rr < 2; rr++) {
        int id = tid + rr * 256;
        int r = id >> 2, c4 = (id & 3) << 2;
        async_cp16(&Bs[buf][r][c4], B + (tn + r) * (long)ldb + k0 + c4);
      }
    } else {  // B tile 16x128
#pragma unroll
      for (int rr = 0; rr < 2; rr++) {
        int id = tid + rr * 256;
        int r = id >> 5, c4 = (id & 31) << 2;
        async_cp16(&Bs[buf][r][c4], B + (long)(k0 + r) * ldb + tn + c4);
      }
    }
  };

  stage(0, 0);
  const int nk = K >> 4;
  for (int t = 0; t < nk; t++) {
    const int cur = t & 1;
    async_wait0();       // this wave's async writes (for tile t) done
    __syncthreads();     // => all waves' writes done; prior reads retired
    if (t + 1 < nk) stage(1 - cur, (t + 1) << 4);  // prefetch next tile
#pragma unroll
    for (int kk = 0; kk < 16; kk += 4) {
      v2f af[2], bf[2];
#pragma unroll
      for (int i = 0; i < 2; i++) {
        int r = wm * 32 + i * 16 + ln16;
        af[i].x = As[cur][r][kk + 2 * hi];
        af[i].y = As[cur][r][kk + 2 * hi + 1];
      }
#pragma unroll
      for (int j = 0; j < 2; j++) {
        int c = wn * 32 + j * 16 + ln16;
        if constexpr (TRANSB) {
          bf[j].x = Bs[cur][c][kk + 2 * hi];
          bf[j].y = Bs[cur][c][kk + 2 * hi + 1];
        } else {
          bf[j].x = Bs[cur][kk + 2 * hi][c];
          bf[j].y = Bs[cur][kk + 2 * hi + 1][c];
        }
      }
#pragma unroll
      for (int i = 0; i < 2; i++)
#pragma unroll
        for (int j = 0; j < 2; j++) acc[i][j] = wmma4(af[i], bf[j], acc[i][j]);
    }
  }
  // Epilogue: C-layout — VGPR v: lanes0-15 row=v, lanes16-31 row=v+8
#pragma unroll
  for (int i = 0; i < 2; i++)
#pragma unroll
    for (int j = 0; j < 2; j++) {
      long col = tn + wn * 32 + j * 16 + ln16;
#pragma unroll
      for (int v = 0; v < 8; v++) {
        long row = tm + wm * 32 + i * 16 + v + hi * 8;
        float x = alpha * acc[i][j][v];
        if (bias) x += bias[col];
        if (resid) x += resid[row * (long)ldc + col];
        if (act == 1)
          x = fmaxf(x, 0.f);
        else if (act == 2)
          x = fminf(fmaxf(x, -50.f), 50.f);
        C[row * (long)ldc + col] = x;
      }
    }
}

// ---------------------------------------------------------------------------
// Flash attention, fp32 WMMA. grid=(Lq/128, H, B); 8 waves x 16-query tiles.
// Q/K/V are (B, L, ld) with per-head 64-col slices at h*64. qbs==0 => Q
// broadcast over batch. Optional bias (B, Lq, Lk): clamp(bscale*bias, +-1e4).
// K/V tiles double-buffered via async global->LDS.
// ---------------------------------------------------------------------------
__global__ __launch_bounds__(256) void k_attn(
    const float* __restrict__ Q, long qbs, int qld, const float* __restrict__ Kp,
    long kbs, int kld, const float* __restrict__ Vp, long vbs, int vld,
    float* __restrict__ O, long obs, int oldc, const float* __restrict__ bias,
    long bbs, float bscale, int Lq, int Lk) {
  __shared__ alignas(16) float Ks[2][32][68];
  __shared__ alignas(16) float Vs[2][32][68];
  __shared__ float Ps[8][16][36];

  const int tid = threadIdx.x, lane = tid & 31, wv = tid >> 5;
  const int hi = (lane >> 4) & 1, ln16 = lane & 15;
  const int h = blockIdx.y, b = blockIdx.z;
  const int hoff = h * 64;
  const int q0 = blockIdx.x * 128 + wv * 16;
  (void)Lq;

  auto stageKV = [&](int buf, int kt) {
#pragma unroll
    for (int rr = 0; rr < 2; rr++) {  // 32 keys x 64 dims each for K and V
      int id = tid + rr * 256;
      int r = id >> 4, c4 = (id & 15) << 2;
      async_cp16(&Ks[buf][r][c4],
                 Kp + (long)b * kbs + (long)(kt + r) * kld + hoff + c4);
      async_cp16(&Vs[buf][r][c4],
                 Vp + (long)b * vbs + (long)(kt + r) * vld + hoff + c4);
    }
  };

  // Preload Q fragments (16 k-steps of depth 4), pre-scaled by 1/sqrt(DK)
  v2f qf[16];
  {
    const float* qrow = Q + (long)b * qbs + (long)(q0 + ln16) * qld + hoff;
#pragma unroll
    for (int i = 0; i < 16; i++) {
      qf[i].x = qrow[i * 4 + 2 * hi] * 0.125f;
      qf[i].y = qrow[i * 4 + 2 * hi + 1] * 0.125f;
    }
  }
  v8f vz = {0.f, 0.f, 0.f, 0.f, 0.f, 0.f, 0.f, 0.f};
  v8f ac[4];
#pragma unroll
  for (int j = 0; j < 4; j++) ac[j] = vz;
  float m8[8], l8[8];
#pragma unroll
  for (int v = 0; v < 8; v++) {
    m8[v] = -3.0e38f;
    l8[v] = 0.f;
  }

  stageKV(0, 0);
  const int nkt = Lk >> 5;
  for (int t = 0; t < nkt; t++) {
    const int cur = t & 1;
    const int kt = t << 5;
    async_wait0();
    __syncthreads();
    if (t + 1 < nkt) stageKV(1 - cur, kt + 32);

    v8f s[2];
#pragma unroll
    for (int st = 0; st < 2; st++) {  // S = Q K^T (two 16x16 tiles)
      s[st] = vz;
#pragma unroll
      for (int i = 0; i < 16; i++) {
        v2f bf;
        int key = st * 16 + ln16;
        bf.x = Ks[cur][key][i * 4 + 2 * hi];
        bf.y = Ks[cur][key][i * 4 + 2 * hi + 1];
        s[st] = wmma4(qf[i], bf, s[st]);
      }
      if (bias) {
#pragma unroll
        for (int v = 0; v < 8; v++) {
          long row = q0 + v + hi * 8;
          long col = kt + st * 16 + ln16;
          float bv = bias[(long)b * bbs + row * (long)Lk + col] * bscale;
          bv = fminf(fmaxf(bv, -1.0e4f), 1.0e4f);
          s[st][v] += bv;
        }
      }
    }
    // online softmax update
    float pmax[8];
#pragma unroll
    for (int v = 0; v < 8; v++) {
      float mx = fmaxf(s[0][v], s[1][v]);
      mx = hmax16(mx);
      pmax[v] = fmaxf(m8[v], mx);
    }
#pragma unroll
    for (int v = 0; v < 8; v++) {
      float sc = __expf(m8[v] - pmax[v]);
      l8[v] *= sc;
      m8[v] = pmax[v];
#pragma unroll
      for (int j = 0; j < 4; j++) ac[j][v] *= sc;
    }
#pragma unroll
    for (int st = 0; st < 2; st++) {
#pragma unroll
      for (int v = 0; v < 8; v++) {
        float e = __expf(s[st][v] - m8[v]);
        s[st][v] = e;
        Ps[wv][v + hi * 8][st * 16 + ln16] = e;  // C-layout -> LDS (A source)
      }
    }
#pragma unroll
    for (int v = 0; v < 8; v++) {
      float rs = s[0][v] + s[1][v];
      rs = hsum16(rs);
      l8[v] += rs;
    }
    // acc += P (16x32) @ V (32x64)
#pragma unroll
    for (int i = 0; i < 8; i++) {
      v2f af;
      af.x = Ps[wv][ln16][i * 4 + 2 * hi];
      af.y = Ps[wv][ln16][i * 4 + 2 * hi + 1];
#pragma unroll
      for (int j = 0; j < 4; j++) {
        v2f bf;
        bf.x = Vs[cur][i * 4 + 2 * hi][j * 16 + ln16];
        bf.y = Vs[cur][i * 4 + 2 * hi + 1][j * 16 + ln16];
        ac[j] = wmma4(af, bf, ac[j]);
      }
    }
  }
#pragma unroll
  for (int v = 0; v < 8; v++) l8[v] = 1.f / l8[v];
#pragma unroll
  for (int j = 0; j < 4; j++)
#pragma unroll
    for (int v = 0; v < 8; v++) {
      long row = q0 + v + hi * 8;
      O[(long)b * obs + row * (long)oldc + hoff + j * 16 + ln16] =
          ac[j][v] * l8[v];
    }
}

// ------------------------- elementwise / LN kernels -------------------------
__global__ __launch_bounds__(256) void k_ln(const float* __restrict__ x,
                                            const float* __restrict__ g,
                                            const float* __restrict__ be,
                                            float* __restrict__ out, int ntok,
                                            float eps) {
  int lane = threadIdx.x & 31, wv = threadIdx.x >> 5;
  long m = (long)blockIdx.x * 8 + wv;
  if (m >= ntok) return;
  const float* row = x + m * DD;
  float v[16];
  float s = 0.f;
#pragma unroll
  for (int j = 0; j < 16; j++) { v[j] = row[lane + j * 32]; s += v[j]; }
  s = wsum32(s);
  float mean = s * (1.f / DD);
  float q = 0.f;
#pragma unroll
  for (int j = 0; j < 16; j++) { float d = v[j] - mean; q += d * d; }
  q = wsum32(q);
  float rstd = rsqrtf(q * (1.f / DD) + eps);
#pragma unroll
  for (int j = 0; j < 16; j++) {
    int d = lane + j * 32;
    out[m * DD + d] = (v[j] - mean) * rstd * g[d] + be[d];
  }
}

__global__ __launch_bounds__(256) void k_ln_res(
    const float* __restrict__ qp, int qmod, const float* __restrict__ a,
    float alpha, const float* __restrict__ g, const float* __restrict__ be,
    float eps, float* __restrict__ out, int ntok) {
  int lane = threadIdx.x & 31, wv = threadIdx.x >> 5;
  long m = (long)blockIdx.x * 8 + wv;
  if (m >= ntok) return;
  long qm = qmod ? (m % qmod) : m;
  float v[16];
  float s = 0.f;
#pragma unroll
  for (int j = 0; j < 16; j++) {
    int d = lane + j * 32;
    v[j] = qp[qm * DD + d] + alpha * a[m * DD + d];
    s += v[j];
  }
  s = wsum32(s);
  float mean = s * (1.f / DD);
  float q = 0.f;
#pragma unroll
  for (int j = 0; j < 16; j++) { float d = v[j] - mean; q += d * d; }
  q = wsum32(q);
  float rstd = rsqrtf(q * (1.f / DD) + eps);
#pragma unroll
  for (int j = 0; j < 16; j++) {
    int d = lane + j * 32;
    out[m * DD + d] = (v[j] - mean) * rstd * g[d] + be[d];
  }
}

// rank-2 factorization of dense_in: P=sum_f max(w1,0)*w2, N=sum_f min(w1,0)*w2
__global__ void k_pn(const float* __restrict__ w1, const float* __restrict__ w2,
                     float* P, float* Nn) {
  int d = blockIdx.x * 256 + threadIdx.x;
  if (d >= DD) return;
  float p = 0.f, n = 0.f;
  for (int f = 0; f < FFD; f++) {
    float w = w1[f];
    float x = w2[(long)f * DD + d];
    p += fmaxf(w, 0.f) * x;
    n += fminf(w, 0.f) * x;
  }
  P[d] = p;
  Nn[d] = n;
}

__global__ __launch_bounds__(256) void k_dense_atac(
    const float* __restrict__ aw, const float* __restrict__ P,
    const float* __restrict__ Nn, const float* __restrict__ g,
    const float* __restrict__ be, const float* __restrict__ pe,
    float* __restrict__ out, int ntok) {
  int lane = threadIdx.x & 31, wv = threadIdx.x >> 5;
  long m = (long)blockIdx.x * 8 + wv;
  if (m >= ntok) return;
  float a = aw[m];
  const float* PN = (a >= 0.f) ? P : Nn;
  long wpos = m % WW;
  float v[16];
  float s = 0.f;
#pragma unroll
  for (int j = 0; j < 16; j++) { v[j] = a * PN[lane + j * 32]; s += v[j]; }
  s = wsum32(s);
  float mean = s * (1.f / DD);
  float q = 0.f;
#pragma unroll
  for (int j = 0; j < 16; j++) { float d = v[j] - mean; q += d * d; }
  q = wsum32(q);
  float rstd = rsqrtf(q * (1.f / DD) + 1e-5f);
#pragma unroll
  for (int j = 0; j < 16; j++) {
    int d = lane + j * 32;
    out[m * DD + d] = (v[j] - mean) * rstd * g[d] + be[d] + pe[wpos * DD + d];
  }
}

__global__ __launch_bounds__(256) void k_dense_tf(
    const float* __restrict__ tfe, const float* __restrict__ P,
    const float* __restrict__ Nn, const float* __restrict__ g,
    const float* __restrict__ be, const float* __restrict__ emb,
    const int* __restrict__ ids, float* __restrict__ out, int ntok) {
  int lane = threadIdx.x & 31, wv = threadIdx.x >> 5;
  long m = (long)blockIdx.x * 8 + wv;
  if (m >= ntok) return;
  float a = tfe[m];
  const float* PN = (a >= 0.f) ? P : Nn;
  long t = m % TT;
  long id = ids[t];
  float v[16];
  float s = 0.f;
#pragma unroll
  for (int j = 0; j < 16; j++) { v[j] = a * PN[lane + j * 32]; s += v[j]; }
  s = wsum32(s);
  float mean = s * (1.f / DD);
  float q = 0.f;
#pragma unroll
  for (int j = 0; j < 16; j++) { float d = v[j] - mean; q += d * d; }
  q = wsum32(q);
  float rstd = rsqrtf(q * (1.f / DD) + 1e-5f);
#pragma unroll
  for (int j = 0; j < 16; j++) {
    int d = lane + j * 32;
    out[m * DD + d] = (v[j] - mean) * rstd * g[d] + be[d] + emb[id * DD + d];
  }
}

__global__ void k_posenc(float* __restrict__ pe) {
  int n = blockIdx.x;
  int d = threadIdx.x;  // 0..255
  float inv = __expf(-((float)(2 * d) / (float)DD) * logf(10000.f));
  float si = (float)n * inv;
  pe[(long)n * DD + d] = sinf(si);
  pe[(long)n * DD + 256 + d] = cosf(si);
}

__global__ void k_gather(const float* __restrict__ emb,
                         const int* __restrict__ ids, float* __restrict__ out,
                         int n) {
  long idx = (long)blockIdx.x * 256 + threadIdx.x;
  if (idx >= (long)n * DD) return;
  long r = idx / DD, c = idx % DD;
  out[idx] = emb[(long)ids[r] * DD + c];
}

__global__ __launch_bounds__(256) void k_attn_pool(const float* __restrict__ x,
                                                   const float* __restrict__ q,
                                                   float* __restrict__ out,
                                                   int L) {
  __shared__ float sc[WW];
  __shared__ float red[256];
  int b = blockIdx.x, tid = threadIdx.x;
  for (int l = tid; l < L; l += 256) {
    const float* row = x + ((long)b * L + l) * DD;
    float s = 0.f;
    for (int d = 0; d < DD; d++) s += row[d] * q[d];
    sc[l] = s * 0.044194173824159216f;  // 1/sqrt(512)
  }
  __syncthreads();
  float mx = -3.0e38f;
  for (int l = tid; l < L; l += 256) mx = fmaxf(mx, sc[l]);
  red[tid] = mx;
  __syncthreads();
  for (int s = 128; s > 0; s >>= 1) {
    if (tid < s) red[tid] = fmaxf(red[tid], red[tid + s]);
    __syncthreads();
  }
  mx = red[0];
  __syncthreads();
  float sm = 0.f;
  for (int l = tid; l < L; l += 256) {
    float e = __expf(sc[l] - mx);
    sc[l] = e;
    sm += e;
  }
  red[tid] = sm;
  __syncthreads();
  for (int s = 128; s > 0; s >>= 1) {
    if (tid < s) red[tid] += red[tid + s];
    __syncthreads();
  }
  float inv = 1.f / red[0];
  __syncthreads();
  for (int d = tid; d < DD; d += 256) {
    float acc = 0.f;
    for (int l = 0; l < L; l++) acc += sc[l] * x[((long)b * L + l) * DD + d];
    out[(long)b * DD + d] = acc * inv;
  }
}

__global__ __launch_bounds__(256) void k_pooled(
    const float* __restrict__ p1, const float* __restrict__ p2,
    const float* __restrict__ w1, const float* __restrict__ w2,
    const float* __restrict__ g, const float* __restrict__ be,
    float* __restrict__ ctx) {
  __shared__ float hc[2 * DD];
  __shared__ float hf[FFD];
  __shared__ float vv[DD];
  __shared__ float red[256];
  int b = blockIdx.x, tid = threadIdx.x;
  for (int d = tid; d < DD; d += 256) {
    hc[d] = p1[(long)b * DD + d];
    hc[DD + d] = p2[(long)b * DD + d];
  }
  __syncthreads();
  for (int f = tid; f < FFD; f += 256) {
    float s = 0.f;
    for (int k = 0; k < 2 * DD; k++) s += hc[k] * w1[(long)k * FFD + f];
    hf[f] = 0.5f * s * (1.f + erff(s * 0.7071067811865475f));  // exact gelu
  }
  __syncthreads();
  for (int d = tid; d < DD; d += 256) {
    float s = 0.f;
    for (int f = 0; f < FFD; f++) s += hf[f] * w2[(long)f * DD + d];
    vv[d] = s;
  }
  __syncthreads();
  float ls = 0.f;
  for (int d = tid; d < DD; d += 256) ls += vv[d];
  red[tid] = ls;
  __syncthreads();
  for (int s = 128; s > 0; s >>= 1) {
    if (tid < s) red[tid] += red[tid + s];
    __syncthreads();
  }
  float mean = red[0] * (1.f / DD);
  __syncthreads();
  float lq = 0.f;
  for (int d = tid; d < DD; d += 256) {
    float dd = vv[d] - mean;
    lq += dd * dd;
  }
  red[tid] = lq;
  __syncthreads();
  for (int s = 128; s > 0; s >>= 1) {
    if (tid < s) red[tid] += red[tid + s];
    __syncthreads();
  }
  float rstd = rsqrtf(red[0] * (1.f / DD) + 1e-5f);
  __syncthreads();
  for (int d = tid; d < DD; d += 256)
    ctx[(long)b * DD + d] = (vv[d] - mean) * rstd * g[d] + be[d];
}

__global__ void k_add_ctx(const float* __restrict__ t,
                          const float* __restrict__ ctx, float* __restrict__ out,
                          long n) {
  long i = (long)blockIdx.x * 256 + threadIdx.x;
  if (i >= n) return;
  long b = i / ((long)GG * DD);
  long d = i % DD;
  out[i] = t[i] + ctx[b * DD + d];
}

__global__ void k_softmax_rows(float* __restrict__ x, int nrows) {
  int lane = threadIdx.x & 31, wv = threadIdx.x >> 5;
  long r = (long)blockIdx.x * 8 + wv;
  if (r >= nrows) return;
  float* row = x + r * TT;
  float v[16];
  float mx = -3.0e38f;
#pragma unroll
  for (int j = 0; j < 16; j++) {
    v[j] = row[lane + 32 * j];
    mx = fmaxf(mx, v[j]);
  }
  mx = wmax32(mx);
  float s = 0.f;
#pragma unroll
  for (int j = 0; j < 16; j++) {
    v[j] = __expf(v[j] - mx);
    s += v[j];
  }
  s = wsum32(s);
  float inv = 1.f / s;
#pragma unroll
  for (int j = 0; j < 16; j++) row[lane + 32 * j] = v[j] * inv;
}

__global__ void k_shortcut(const float* __restrict__ tfe,
                           const float* __restrict__ attn,
                           float* __restrict__ scv, int n) {
  int i = blockIdx.x * 256 + threadIdx.x;
  if (i >= n) return;
  int b = i / GG, g = i % GG;
  float s = 0.f;
  for (int t = 0; t < TT; t++) s += tfe[(long)b * TT + t] * attn[(long)g * TT + t];
  scv[i] = s;
}

__global__ void k_pred_final(const float* __restrict__ big,
                             const float* __restrict__ w2,
                             const float* __restrict__ scv,
                             const float* __restrict__ scale,
                             float* __restrict__ out, int ntok) {
  int lane = threadIdx.x & 31, wv = threadIdx.x >> 5;
  long m = (long)blockIdx.x * 8 + wv;
  if (m >= ntok) return;
  const float* row = big + m * FFD;
  float s = 0.f;
  for (int j = 0; j < 64; j++) {
    int f = lane + 32 * j;
    s += row[f] * w2[f];
  }
  s = wsum32(s);
  if (lane == 0) out[m] = s + scale[0] * scv[m];
}

// ---------------------------------------------------------------------------
extern "C" void kernel_launch(void* const* d_in, const int* in_sizes, int n_in,
                              void* d_out, int out_size, void* d_ws,
                              size_t ws_size, hipStream_t stream) {
  (void)in_sizes; (void)n_in; (void)out_size; (void)ws_size;
  const float* atac_w = (const float*)d_in[0];
  const float* tf_expr = (const float*)d_in[1];
  const float* bias = (const float*)d_in[2];
  auto F = [&](int i) { return (const float*)d_in[i]; };
  const int* tf_ids = (const int*)d_in[101];
  const int* tg_ids = (const int*)d_in[102];
  float* out = (float*)d_out;

  float* w = (float*)d_ws;
  size_t off = 0;
  auto alloc = [&](size_t n) { float* p = w + off; off += n; return p; };
  float* PE = alloc((size_t)WW * DD);
  float* AP = alloc(DD);
  float* AN = alloc(DD);
  float* TP = alloc(DD);
  float* TN = alloc(DD);
  float* CTX = alloc((size_t)BB * DD);
  float* P1 = alloc((size_t)BB * DD);
  float* P2 = alloc((size_t)BB * DD);
  float* SC = alloc((size_t)BB * GG);
  float* TFE = alloc((size_t)TT * DD);
  float* TGE = alloc((size_t)GG * DD);
  float* SIM = alloc((size_t)GG * TT);
  float* TGQ0 = alloc((size_t)GG * DD);
  float* TGQ = alloc((size_t)GG * DD);
  float* TFT = alloc((size_t)BB * TT * DD);
  float* TF2A = alloc((size_t)BB * TT * DD);
  float* X = alloc((size_t)BB * WW * DD);
  float* Hb = alloc((size_t)BB * WW * DD);
  float* AO = alloc((size_t)BB * WW * DD);
  float* QB = alloc((size_t)BB * WW * DD);
  float* KB = alloc((size_t)BB * WW * DD);
  float* VB = alloc((size_t)BB * WW * DD);
  float* BIG = alloc((size_t)BB * WW * FFD);  // qkv / FF / pred-hidden

  auto gemm = [&](const float* A, const float* Bm, const float* bi,
                  const float* rs, float* C, int M, int N, int K, float alpha,
                  int act) {
    dim3 grid(N / 128, M / 64);
    k_gemm<false><<<grid, 256, 0, stream>>>(A, Bm, bi, rs, C, M, N, K, K, N, N,
                                            alpha, act);
  };
  auto gemmT = [&](const float* A, const float* Bm, const float* bi,
                   const float* rs, float* C, int M, int N, int K, float alpha,
                   int act) {
    dim3 grid(N / 128, M / 64);
    k_gemm<true><<<grid, 256, 0, stream>>>(A, Bm, bi, rs, C, M, N, K, K, K, N,
                                           alpha, act);
  };

  // --- input stems ---
  k_posenc<<<WW, 256, 0, stream>>>(PE);
  k_pn<<<2, 256, 0, stream>>>(F(10), F(11), AP, AN);  // atacin
  k_pn<<<2, 256, 0, stream>>>(F(6), F(7), TP, TN);    // tfin
  k_dense_atac<<<BB * WW / 8, 256, 0, stream>>>(atac_w, AP, AN, F(12), F(13),
                                                PE, X, BB * WW);

  // --- encoder x4 (norm_first) ---
  for (int l = 0; l < 4; l++) {
    int p = 23 + l * 12;
    k_ln<<<BB * WW / 8, 256, 0, stream>>>(X, F(p + 8), F(p + 9), Hb, BB * WW,
                                          1e-5f);
    gemm(Hb, F(p + 0), F(p + 1), nullptr, BIG, BB * WW, 3 * DD, DD, 1.f, 0);
    dim3 ag(WW / 128, HH, BB);
    k_attn<<<ag, 256, 0, stream>>>(BIG, (long)WW * 3 * DD, 3 * DD, BIG + DD,
                                   (long)WW * 3 * DD, 3 * DD, BIG + 2 * DD,
                                   (long)WW * 3 * DD, 3 * DD, AO, (long)WW * DD,
                                   DD, nullptr, 0, 0.f, WW, WW);
    gemm(AO, F(p + 2), F(p + 3), X, X, BB * WW, DD, DD, 1.f, 0);
    k_ln<<<BB * WW / 8, 256, 0, stream>>>(X, F(p + 10), F(p + 11), Hb, BB * WW,
                                          1e-5f);
    gemm(Hb, F(p + 4), F(p + 5), nullptr, BIG, BB * WW, FFD, DD, 1.f, 1);
    gemm(BIG, F(p + 6), F(p + 7), X, X, BB * WW, DD, FFD, 1.f, 0);
  }

  // --- TF token stream ---
  k_dense_tf<<<BB * TT / 8, 256, 0, stream>>>(tf_expr, TP, TN, F(8), F(9),
                                              F(3), tf_ids, TFT, BB * TT);

  // --- tf2atac cross attention ---
  {
    int p = 71;
    gemm(TFT, F(p + 0), F(p + 1), nullptr, QB, BB * TT, DD, DD, 1.f, 0);
    gemm(X, F(p + 2), F(p + 3), nullptr, KB, BB * WW, DD, DD, 1.f, 0);
    gemm(X, F(p + 4), F(p + 5), nullptr, VB, BB * WW, DD, DD, 1.f, 0);
    dim3 ag(TT / 128, HH, BB);
    k_attn<<<ag, 256, 0, stream>>>(QB, (long)TT * DD, DD, KB, (long)WW * DD, DD,
                                   VB, (long)WW * DD, DD, AO, (long)TT * DD, DD,
                                   nullptr, 0, 0.f, TT, WW);
    gemm(AO, F(p + 6), F(p + 7), nullptr, BIG, BB * TT, DD, DD, 1.f, 0);
    k_ln_res<<<BB * TT / 8, 256, 0, stream>>>(TFT, 0, BIG, 0.1f, F(p + 8),
                                              F(p + 9), 1e-6f, TF2A, BB * TT);
  }
  // --- atac2tf cross attention ---
  {
    int p = 81;
    gemm(X, F(p + 0), F(p + 1), nullptr, QB, BB * WW, DD, DD, 1.f, 0);
    gemm(TFT, F(p + 2), F(p + 3), nullptr, KB, BB * TT, DD, DD, 1.f, 0);
    gemm(TFT, F(p + 4), F(p + 5), nullptr, VB, BB * TT, DD, DD, 1.f, 0);
    dim3 ag(WW / 128, HH, BB);
    k_attn<<<ag, 256, 0, stream>>>(QB, (long)WW * DD, DD, KB, (long)TT * DD, DD,
                                   VB, (long)TT * DD, DD, AO, (long)WW * DD, DD,
                                   nullptr, 0, 0.f, WW, TT);
    gemm(AO, F(p + 6), F(p + 7), nullptr, BIG, BB * WW, DD, DD, 1.f, 0);
    k_ln_res<<<BB * WW / 8, 256, 0, stream>>>(X, 0, BIG, 0.1f, F(p + 8),
                                              F(p + 9), 1e-6f, Hb, BB * WW);
  }

  // --- pooling + context MLP ---
  k_attn_pool<<<BB, 256, 0, stream>>>(TF2A, F(14), P1, TT);
  k_attn_pool<<<BB, 256, 0, stream>>>(Hb, F(15), P2, WW);
  k_pooled<<<BB, 256, 0, stream>>>(P1, P2, F(16), F(17), F(18), F(19), CTX);

  // --- gathers ---
  k_gather<<<(GG * DD) / 256, 256, 0, stream>>>(F(4), tg_ids, TGQ0, GG);
  k_gather<<<(GG * DD) / 256, 256, 0, stream>>>(F(5), tg_ids, TGE, GG);
  k_gather<<<(TT * DD) / 256, 256, 0, stream>>>(F(3), tf_ids, TFE, TT);

  // --- tg2atac cross attention (biased, broadcast Q) ---
  {
    int p = 91;
    gemm(TGQ0, F(p + 0), F(p + 1), nullptr, TGQ, GG, DD, DD, 1.f, 0);
    gemm(X, F(p + 2), F(p + 3), nullptr, KB, BB * WW, DD, DD, 1.f, 0);
    gemm(X, F(p + 4), F(p + 5), nullptr, VB, BB * WW, DD, DD, 1.f, 0);
    dim3 ag(GG / 128, HH, BB);
    k_attn<<<ag, 256, 0, stream>>>(TGQ, 0L, DD, KB, (long)WW * DD, DD, VB,
                                   (long)WW * DD, DD, AO, (long)GG * DD, DD,
                                   bias, (long)GG * WW, 2.0f, GG, WW);
    gemm(AO, F(p + 6), F(p + 7), nullptr, BIG, BB * GG, DD, DD, 1.f, 0);
    k_ln_res<<<BB * GG / 8, 256, 0, stream>>>(TGQ0, GG, BIG, 0.1f, F(p + 8),
                                              F(p + 9), 1e-6f, KB, BB * GG);
  }

  // --- prediction head + shortcut ---
  k_add_ctx<<<((long)BB * GG * DD) / 256, 256, 0, stream>>>(KB, CTX, VB,
                                                            (long)BB * GG * DD);
  gemm(VB, F(20), nullptr, nullptr, BIG, BB * GG, FFD, DD, 1.f, 1);
  gemmT(TGE, TFE, nullptr, nullptr, SIM, GG, TT, DD, 0.044194173824159216f, 2);
  k_softmax_rows<<<GG / 8, 256, 0, stream>>>(SIM, GG);
  k_shortcut<<<(BB * GG) / 256, 256, 0, stream>>>(tf_expr, SIM, SC, BB * GG);
  k_pred_final<<<BB * GG / 8, 256, 0, stream>>>(BIG, F(21), SC, F(22), out,
                                                BB * GG);
}